// GNN_57638461112420
// MI455X (gfx1250) — compile-verified
//
#include <hip/hip_runtime.h>
#include <math.h>

// GATConv forward for MI455X (gfx1250).
// Heavy parts: L2-resident scatter atomics; dense projection h = x@W done with
// V_WMMA_F32_16X16X4_F32 (exact f32, K padded 3->4).

#define H_HEADS 4
#define C_CH    32
#define HC      128   // H_HEADS * C_CH
#define SLOPE   0.2f

typedef __attribute__((ext_vector_type(2))) float v2f;
typedef __attribute__((ext_vector_type(8))) float v8f;

__device__ __forceinline__ float leaky(float a) { return a > 0.0f ? a : SLOPE * a; }

// Order-preserving float <-> uint mapping so atomicMax(uint) == float max.
__device__ __forceinline__ unsigned f2ord(float f) {
  unsigned u = __float_as_uint(f);
  return (u & 0x80000000u) ? ~u : (u | 0x80000000u);
}
__device__ __forceinline__ float ord2f(unsigned o) {
  unsigned u = (o & 0x80000000u) ? (o & 0x7FFFFFFFu) : ~o;
  return __uint_as_float(u);
}

// ---------------- zero scratch (deg + esum) ----------------
__global__ void k_zero(unsigned* __restrict__ p, int n) {
  int i = blockIdx.x * blockDim.x + threadIdx.x;
  if (i < n) p[i] = 0u;
}

// ---------------- per-head edge-attention constant k[h] = dot(W_edge[h], att_edge[h]) ----
__global__ void k_consts(const float* __restrict__ W_edge, const float* __restrict__ att_edge,
                         float* __restrict__ kv) {
  int hh = threadIdx.x;
  if (hh < H_HEADS) {
    float s = 0.0f;
    #pragma unroll
    for (int c = 0; c < C_CH; ++c) s += W_edge[hh * C_CH + c] * att_edge[hh * C_CH + c];
    kv[hh] = s;
  }
}

// ---------------- edge pass 0: in-degree + sum of edge attrs per dst ----------------
__global__ void __launch_bounds__(256) k_deg(const int* __restrict__ dst, const float* __restrict__ ef,
                                             int* __restrict__ deg, float* __restrict__ esum, int E) {
  int e = blockIdx.x * blockDim.x + threadIdx.x;
  if (e >= E) return;
  int d = dst[e];
  atomicAdd(&deg[d], 1);
  unsafeAtomicAdd(&esum[d], ef[e]);
}

// ---------------- h = x_pad[Nx4] @ W_pad[4x128] via V_WMMA_F32_16X16X4_F32 ----------------
// One wave computes a 16-row stripe of h (8 WMMA tiles across the 128 columns).
// A/B fragments built branchlessly (v_cndmask, no EXEC churn); stores take a
// wave-uniform full-tile fast path with straight-line coalesced b32 stores.
__global__ void __launch_bounds__(256) k_h_wmma(const float* __restrict__ x,  // [N,3]
                                                const float* __restrict__ W,  // [3,128]
                                                float* __restrict__ h,        // [N,128]
                                                int N, int nTiles) {
  int wave = (int)((blockIdx.x * blockDim.x + threadIdx.x) >> 5);
  int lane = threadIdx.x & 31;
  if (wave >= nTiles) return;              // wave-uniform: EXEC stays all-ones for WMMA
  int row0 = wave * 16;
  int half = lane >> 4;                    // 0 => K=0,1   1 => K=2,3
  int lm   = lane & 15;
  bool hi  = (half != 0);
  bool full = (row0 + 16 <= N);            // wave-uniform

  // A (16x4 f32): lanes 0-15 VGPR{0,1} = K{0,1}; lanes 16-31 = K{2,3}. K=3 padded 0.
  int arow = row0 + lm; if (arow >= N) arow = N - 1;   // clamp; writes are guarded in tail
  const float* xr = x + (size_t)arow * 3;
  float x0 = xr[0], x1 = xr[1], x2 = xr[2];
  v2f a;
  a.x = hi ? x2 : x0;
  a.y = hi ? 0.0f : x1;

  #pragma unroll
  for (int j = 0; j < 8; ++j) {
    int col = j * 16 + lm;
    // B (4x16 f32): lanes 0-15 VGPR{0,1} = K{0,1}; lanes 16-31 = K{2,3}. K=3 padded 0.
    float b0 = W[col], b1 = W[HC + col], b2 = W[2 * HC + col];
    v2f b;
    b.x = hi ? b2 : b0;
    b.y = hi ? 0.0f : b1;
    v8f c = {};
    c = __builtin_amdgcn_wmma_f32_16x16x4_f32(false, a, false, b, (short)0, c, false, false);
    // D layout: lane lm = column; VGPR v = row v (lanes 0-15) / row v+8 (lanes 16-31).
    float* hb = h + ((size_t)row0 + (size_t)half * 8) * HC + col;
    if (full) {
      #pragma unroll
      for (int v = 0; v < 8; ++v) hb[(size_t)v * HC] = c[v];
    } else {
      #pragma unroll
      for (int v = 0; v < 8; ++v) {
        if (row0 + half * 8 + v < N) hb[(size_t)v * HC] = c[v];
      }
    }
  }
}

// ---------------- per (node,head): a_src, a_dst, self-loop logit, init segment-max ------
__global__ void __launch_bounds__(256) k_node_att(const float* __restrict__ h,
                                                  const float* __restrict__ att_src,
                                                  const float* __restrict__ att_dst,
                                                  const int* __restrict__ deg,
                                                  const float* __restrict__ esum,
                                                  const float* __restrict__ kv,
                                                  float* __restrict__ a_src, float* __restrict__ a_dst,
                                                  float* __restrict__ alpha_self,
                                                  unsigned* __restrict__ m, int NH) {
  int t = blockIdx.x * blockDim.x + threadIdx.x;
  if (t >= NH) return;
  int n = t >> 2, hh = t & 3;
  const float* hp = h + (size_t)n * HC + hh * C_CH;
  const float* as = att_src + hh * C_CH;
  const float* ad = att_dst + hh * C_CH;
  float s = 0.0f, dsum = 0.0f;
  #pragma unroll
  for (int c = 0; c < C_CH; ++c) { float v = hp[c]; s += v * as[c]; dsum += v * ad[c]; }
  a_src[t] = s;
  a_dst[t] = dsum;
  float dg = (float)deg[n]; if (dg < 1.0f) dg = 1.0f;
  float la = esum[n] / dg;                         // self-loop edge attr (mean fill)
  float al = leaky(s + dsum + la * kv[hh]);
  alpha_self[t] = al;
  m[t] = f2ord(al);                                // self-loop guarantees non-empty segment
}

// ---------------- edge pass 1: segment max of logits ----------------
__global__ void __launch_bounds__(256) k_edge_max(const int* __restrict__ src, const int* __restrict__ dst,
                                                  const float* __restrict__ ef, const float* __restrict__ kv,
                                                  const float* __restrict__ a_src, const float* __restrict__ a_dst,
                                                  unsigned* __restrict__ m, int EH) {
  int t = blockIdx.x * blockDim.x + threadIdx.x;
  if (t >= EH) return;
  int e = t >> 2, hh = t & 3;
  int s = src[e], d = dst[e];
  float al = leaky(a_src[s * 4 + hh] + a_dst[d * 4 + hh] + ef[e] * kv[hh]);
  atomicMax(&m[d * 4 + hh], f2ord(al));
}

// ---------------- per (node,head): convert m to float, seed den/out_acc with self-loop ---
__global__ void __launch_bounds__(256) k_node_init(const float* __restrict__ h,
                                                   const float* __restrict__ alpha_self,
                                                   float* __restrict__ m_f,     // in: ordered uint, out: float
                                                   float* __restrict__ den,
                                                   float* __restrict__ out_acc, int NH) {
  int t = blockIdx.x * blockDim.x + threadIdx.x;
  if (t >= NH) return;
  float mm = ord2f(((const unsigned*)m_f)[t]);
  m_f[t] = mm;
  float ex = expf(alpha_self[t] - mm);
  den[t] = ex;
  int n = t >> 2, hh = t & 3;
  const float* hp = h + (size_t)n * HC + hh * C_CH;
  float* op = out_acc + (size_t)n * HC + hh * C_CH;
  #pragma unroll
  for (int c = 0; c < C_CH; ++c) op[c] = ex * hp[c];
}

// ---------------- edge pass 2: exp weights + scatter-add messages (hot loop) ------------
__global__ void __launch_bounds__(256) k_edge_agg(const int* __restrict__ src, const int* __restrict__ dst,
                                                  const float* __restrict__ ef, const float* __restrict__ kv,
                                                  const float* __restrict__ a_src, const float* __restrict__ a_dst,
                                                  const float* __restrict__ m_f,
                                                  const float* __restrict__ h,
                                                  float* __restrict__ den, float* __restrict__ out_acc, int EH) {
  int t = blockIdx.x * blockDim.x + threadIdx.x;
  if (t >= EH) return;
  int e = t >> 2, hh = t & 3;
  int s = src[e], d = dst[e];
  float al = leaky(a_src[s * 4 + hh] + a_dst[d * 4 + hh] + ef[e] * kv[hh]);
  float ex = expf(al - m_f[d * 4 + hh]);
  unsafeAtomicAdd(&den[d * 4 + hh], ex);
  const float4* hp = (const float4*)(h + (size_t)s * HC + hh * C_CH);   // 16B aligned
  float* op = out_acc + (size_t)d * HC + hh * C_CH;
  #pragma unroll
  for (int q = 0; q < 8; ++q) {
    float4 v = hp[q];
    unsafeAtomicAdd(&op[q * 4 + 0], ex * v.x);
    unsafeAtomicAdd(&op[q * 4 + 1], ex * v.y);
    unsafeAtomicAdd(&op[q * 4 + 2], ex * v.z);
    unsafeAtomicAdd(&op[q * 4 + 3], ex * v.w);
  }
}

// ---------------- final: normalize, head-mean, bias ----------------
__global__ void __launch_bounds__(256) k_final(const float* __restrict__ out_acc,
                                               const float* __restrict__ den,
                                               const float* __restrict__ bias,
                                               float* __restrict__ out, int NC) {
  int t = blockIdx.x * blockDim.x + threadIdx.x;
  if (t >= NC) return;
  int n = t >> 5, c = t & 31;
  float s = 0.0f;
  #pragma unroll
  for (int hh = 0; hh < H_HEADS; ++hh)
    s += out_acc[(size_t)n * HC + hh * C_CH + c] / (den[n * 4 + hh] + 1e-16f);
  out[t] = 0.25f * s + bias[c];
}

static inline int ceil_div(long long a, int b) { return (int)((a + b - 1) / b); }

extern "C" void kernel_launch(void* const* d_in, const int* in_sizes, int n_in,
                              void* d_out, int out_size, void* d_ws, size_t ws_size,
                              hipStream_t stream) {
  const float* x        = (const float*)d_in[0];
  const int*   edges    = (const int*)  d_in[1];
  const float* ef       = (const float*)d_in[2];
  const float* W        = (const float*)d_in[3];
  const float* W_edge   = (const float*)d_in[4];
  const float* att_src  = (const float*)d_in[5];
  const float* att_dst  = (const float*)d_in[6];
  const float* att_edge = (const float*)d_in[7];
  const float* bias     = (const float*)d_in[8];
  float* out = (float*)d_out;

  const int N = in_sizes[0] / 3;
  const int E = in_sizes[1] / 2;
  const int* src = edges;
  const int* dst = edges + E;

  // workspace layout (floats)
  float* w = (float*)d_ws;
  size_t off = 0;
  float*    h          = w + off; off += (size_t)N * HC;
  float*    out_acc    = w + off; off += (size_t)N * HC;
  float*    a_src      = w + off; off += (size_t)N * H_HEADS;
  float*    a_dst      = w + off; off += (size_t)N * H_HEADS;
  float*    alpha_self = w + off; off += (size_t)N * H_HEADS;
  float*    m_buf      = w + off; off += (size_t)N * H_HEADS;   // uint during max phase
  float*    den        = w + off; off += (size_t)N * H_HEADS;
  int*      deg        = (int*)(w + off); off += (size_t)N;
  float*    esum       = w + off; off += (size_t)N;
  float*    kv         = w + off; off += 16;
  (void)off; (void)ws_size; (void)n_in; (void)out_size;

  const int B = 256;
  const int NH = N * H_HEADS;
  const long long EH = (long long)E * H_HEADS;
  const int nTiles = (N + 15) / 16;

  // deg + esum are contiguous (2N words): zero them each call
  k_zero<<<ceil_div(2LL * N, B), B, 0, stream>>>((unsigned*)deg, 2 * N);
  k_consts<<<1, 32, 0, stream>>>(W_edge, att_edge, kv);
  k_deg<<<ceil_div(E, B), B, 0, stream>>>(dst, ef, deg, esum, E);
  k_h_wmma<<<ceil_div((long long)nTiles * 32, B), B, 0, stream>>>(x, W, h, N, nTiles);
  k_node_att<<<ceil_div(NH, B), B, 0, stream>>>(h, att_src, att_dst, deg, esum, kv,
                                                a_src, a_dst, alpha_self, (unsigned*)m_buf, NH);
  k_edge_max<<<ceil_div(EH, B), B, 0, stream>>>(src, dst, ef, kv, a_src, a_dst,
                                                (unsigned*)m_buf, (int)EH);
  k_node_init<<<ceil_div(NH, B), B, 0, stream>>>(h, alpha_self, m_buf, den, out_acc, NH);
  k_edge_agg<<<ceil_div(EH, B), B, 0, stream>>>(src, dst, ef, kv, a_src, a_dst, m_buf,
                                                h, den, out_acc, (int)EH);
  k_final<<<ceil_div((long long)N * C_CH, B), B, 0, stream>>>(out_acc, den, bias, out, N * C_CH);
}